// MultiSegmentLoss_76038101008703
// MI455X (gfx1250) — compile-verified
//
#include <hip/hip_runtime.h>

#define NUM_CLASSES 22
#define KANCH 100000
#define NBATCH 8
#define NTGT 32
#define CLIPF 256.0f
#define MAXNF 512.0f
#define EPSF 1.1920929e-07f

typedef float vfloat2 __attribute__((ext_vector_type(2)));

// ---- gfx1250 async global->LDS staging (guarded; falls back to plain LDS store) ----
#if defined(__has_builtin)
# if __has_builtin(__builtin_amdgcn_global_load_async_to_lds_b32) && __has_builtin(__builtin_amdgcn_s_wait_asynccnt)
#  define ASYNC_OK 1
# endif
#endif
#ifndef ASYNC_OK
# define ASYNC_OK 0
#endif

typedef __attribute__((address_space(1))) int* gint_ptr;
typedef __attribute__((address_space(3))) int* lint_ptr;

__device__ __forceinline__ void stage_targets(const float* __restrict__ targets,
                                              int b, float* sT, int tid) {
#if ASYNC_OK
  if (tid < 3 * NTGT) {
    __builtin_amdgcn_global_load_async_to_lds_b32(
        (gint_ptr)(targets + (size_t)b * 3 * NTGT + tid),
        (lint_ptr)(&sT[tid]),
        0, 0);
  }
  __builtin_amdgcn_s_wait_asynccnt(0);
#else
  if (tid < 3 * NTGT) sT[tid] = targets[(size_t)b * 3 * NTGT + tid];
#endif
}

// wave32 butterfly reductions (lower to ds_swizzle / v_permlane on gfx1250)
__device__ __forceinline__ float wred_add(float v) {
#pragma unroll
  for (int off = 16; off > 0; off >>= 1) v += __shfl_xor(v, off, 32);
  return v;
}
__device__ __forceinline__ float wred_max(float v) {
#pragma unroll
  for (int off = 16; off > 0; off >>= 1) v = fmaxf(v, __shfl_xor(v, off, 32));
  return v;
}

// feature-pyramid bounds via cndmask chains (no per-lane memory loads)
__device__ __forceinline__ float bound_lo(int l) {
  float r = 256.0f;
  r = (l == 4) ? 96.0f : r;
  r = (l == 3) ? 60.0f : r;
  r = (l == 2) ? 30.0f : r;
  r = (l == 1) ? 15.0f : r;
  r = (l == 0) ? 0.0f  : r;
  return r;
}
__device__ __forceinline__ float bound_hi(int l) {
  float r = 768.0f;
  r = (l == 3) ? 240.0f : r;
  r = (l == 2) ? 120.0f : r;
  r = (l == 1) ? 60.0f  : r;
  r = (l == 0) ? 30.0f  : r;
  return r;
}

// match(): first-occurrence argmin over 32 targets (matches jnp.argmin tie-break)
__device__ __forceinline__ void match_one(float pc, int lvl, const float* sT,
                                          float& best, int& bi) {
  const float lb = bound_lo(lvl), rb = bound_hi(lvl);
  best = 3.402823466e+38f; bi = 0;
#pragma unroll 8
  for (int n = 0; n < NTGT; ++n) {
    float ts = sT[3 * n], te = sT[3 * n + 1];
    float left  = (pc - ts) * CLIPF;
    float right = (te - pc) * CLIPF;
    float md    = fmaxf(left, right);
    float area  = left + right;
    bool bad = (left < 0.0f) || (right < 0.0f) || (md <= lb) || (md > rb);
    area = bad ? MAXNF : area;
    if (area < best) { best = area; bi = n; }
  }
}

// focal term for one anchor: -alpha_t * (1-pt)^2 * log(pt), pt = softmax[tgt] + 1e-10
// tgt-gather done with a cndmask chain so x[] stays entirely in VGPRs (no scratch).
__device__ __forceinline__ float focal_elem(const float* __restrict__ rowBase, int tgt) {
  const vfloat2* row = (const vfloat2*)rowBase;  // 88B rows, 8B aligned
  float x[NUM_CLASSES];
#pragma unroll
  for (int i = 0; i < NUM_CLASSES / 2; ++i) {
    vfloat2 v = __builtin_nontemporal_load(&row[i]);  // streamed once: NT hint
    x[2 * i] = v.x; x[2 * i + 1] = v.y;
  }
  float m = x[0];
#pragma unroll
  for (int i = 1; i < NUM_CLASSES; ++i) m = fmaxf(m, x[i]);
  float s = 0.0f;
  float xt = x[0];
#pragma unroll
  for (int i = 0; i < NUM_CLASSES; ++i) {
    s += __expf(x[i] - m);
    if (i > 0) xt = (tgt == i) ? x[i] : xt;   // register select, no dynamic index
  }
  float pt = __expf(xt - m) / s + 1e-10f;
  float alpha = (tgt == 0) ? 0.25f : 0.75f;
  float om = 1.0f - pt;
  return -alpha * om * om * __logf(pt);
}

// ws accumulator layout per batch b (8 floats):
// 0: pos_cnt  1: max_pos_iou (f32 bits, all >=0 so int-max == float-max)
// 2: sum_loss_l  3: sum_loss_ct  4: sum_loss_c
// 5: prop_cnt  6: sum_prop_l  7: sum_prop_c
__global__ void msl_init(float* __restrict__ acc) {
  if (threadIdx.x < NBATCH * 8) acc[threadIdx.x] = 0.0f;
}

__global__ __launch_bounds__(256) void msl_pass1(
    const float* __restrict__ loc, const float* __restrict__ conf,
    const float* __restrict__ prop_loc, const float* __restrict__ center,
    const float* __restrict__ priors, const float* __restrict__ targets,
    float* __restrict__ acc) {
  const int b = blockIdx.y;
  const int k = blockIdx.x * 256 + (int)threadIdx.x;
  __shared__ float sT[3 * NTGT];
  __shared__ float sSum[4];   // pos_cnt, loss_l, loss_ct, loss_c
  __shared__ int   sMax;
  if (threadIdx.x < 4) sSum[threadIdx.x] = 0.0f;
  if (threadIdx.x == 0) sMax = 0;
  stage_targets(targets, b, sT, (int)threadIdx.x);
  __syncthreads();

  float v_cnt = 0.f, v_l = 0.f, v_ct = 0.f, v_foc = 0.f, v_iou = 0.f;

  if (k < KANCH) {
    const size_t bk = (size_t)b * KANCH + (size_t)k;
    const float* crow = conf + bk * NUM_CLASSES;
    __builtin_prefetch(crow, 0, 0);  // global_prefetch_b8: pull row while matching

    vfloat2 pr = *(const vfloat2*)(priors + 2 * (size_t)k);
    float pc = pr.x;
    int lvl = (int)pr.y;
    float best; int bi;
    match_one(pc, lvl, sT, best, bi);

    float gs = sT[3 * bi], ge = sT[3 * bi + 1];
    float t0 = (pc - gs) * CLIPF;
    float t1 = (ge - pc) * CLIPF;
    int cf = (best >= MAXNF) ? 0 : (int)sT[3 * bi + 2];

    vfloat2 p = *(const vfloat2*)(loc + bk * 2);
    float inter = fminf(p.x, t0) + fminf(p.y, t1);
    float uni   = p.x + p.y + t0 + t1 - inter;
    float iou   = inter / fmaxf(uni, EPSF);

    if (cf > 0) {
      v_cnt = 1.0f;
      v_iou = iou;
      // GIoU loss
      float ac = fmaxf(p.x, t0) + fmaxf(p.y, t1);
      float gi = iou - (ac - uni) / fmaxf(ac, EPSF);
      v_l = 1.0f - gi;
      // centerness BCE with decoded proposal IoU
      vfloat2 pl = *(const vfloat2*)(prop_loc + bk * 2);
      float pw = p.x + p.y;
      float c0 = 0.5f * pw * pl.x + p.x;
      float c1 = 0.5f * pw * pl.y + p.y;
      float i2 = fminf(c0, t0) + fminf(c1, t1);
      float u2 = c0 + c1 + t0 + t1 - i2;
      float iou2 = fmaxf(i2 / fmaxf(u2, EPSF), 0.0f);
      float xv = center[bk];
      float sp = fmaxf(xv, 0.0f) + log1pf(__expf(-fabsf(xv)));  // stable softplus
      v_ct = sp - xv * iou2;
    }
    v_foc = focal_elem(crow, cf);  // focal over ALL anchors
  }

  // wave32 tree reduction, then one LDS atomic per wave per accumulator
  v_cnt = wred_add(v_cnt);
  v_l   = wred_add(v_l);
  v_ct  = wred_add(v_ct);
  v_foc = wred_add(v_foc);
  v_iou = wred_max(v_iou);
  if ((threadIdx.x & 31) == 0) {
    atomicAdd(&sSum[0], v_cnt);
    atomicAdd(&sSum[1], v_l);
    atomicAdd(&sSum[2], v_ct);
    atomicAdd(&sSum[3], v_foc);
    atomicMax(&sMax, __float_as_int(v_iou));
  }
  __syncthreads();
  if (threadIdx.x == 0) {
    atomicAdd(&acc[b * 8 + 0], sSum[0]);
    atomicMax((int*)&acc[b * 8 + 1], sMax);
    atomicAdd(&acc[b * 8 + 2], sSum[1]);
    atomicAdd(&acc[b * 8 + 3], sSum[2]);
    atomicAdd(&acc[b * 8 + 4], sSum[3]);
  }
}

__global__ __launch_bounds__(256) void msl_pass2(
    const float* __restrict__ loc, const float* __restrict__ prop_loc,
    const float* __restrict__ pconf, const float* __restrict__ priors,
    const float* __restrict__ targets, float* __restrict__ acc) {
  const int b = blockIdx.y;
  const int k = blockIdx.x * 256 + (int)threadIdx.x;
  __shared__ float sT[3 * NTGT];
  __shared__ float sSum[3];   // prop_cnt, prop_l, prop_c
  __shared__ float sThr;
  if (threadIdx.x < 3) sSum[threadIdx.x] = 0.0f;
  if (threadIdx.x == 0) {
    float cnt = acc[b * 8 + 0];
    float mx  = acc[b * 8 + 1];           // stored as f32 bit pattern, >= 0
    float m   = (cnt > 0.0f) ? mx : 2.0f; // has_pos ? max_iou : 2.0
    sThr = fminf(0.5f, m);
  }
  stage_targets(targets, b, sT, (int)threadIdx.x);
  __syncthreads();

  float v_cnt = 0.f, v_sl1 = 0.f, v_foc = 0.f;
  const float thr = sThr;

  if (k < KANCH) {
    const size_t bk = (size_t)b * KANCH + (size_t)k;
    const float* prow = pconf + bk * NUM_CLASSES;
    __builtin_prefetch(prow, 0, 0);

    vfloat2 pr = *(const vfloat2*)(priors + 2 * (size_t)k);
    float pc = pr.x;
    int lvl = (int)pr.y;
    float best; int bi;
    match_one(pc, lvl, sT, best, bi);

    float gs = sT[3 * bi], ge = sT[3 * bi + 1];
    float t0 = (pc - gs) * CLIPF;
    float t1 = (ge - pc) * CLIPF;
    int cf = (best >= MAXNF) ? 0 : (int)sT[3 * bi + 2];

    vfloat2 p = *(const vfloat2*)(loc + bk * 2);
    float inter = fminf(p.x, t0) + fminf(p.y, t1);
    float uni   = p.x + p.y + t0 + t1 - inter;
    float iou   = inter / fmaxf(uni, EPSF);

    int pcf = (iou < thr) ? 0 : cf;
    if (pcf > 0) {
      v_cnt = 1.0f;
      float pw = p.x + p.y;
      float hw = 0.5f * pw;
      float pt0 = (t0 - p.x) / hw;
      float pt1 = (t1 - p.y) / hw;
      vfloat2 pl = *(const vfloat2*)(prop_loc + bk * 2);
      float d0 = fabsf(pl.x - pt0), d1 = fabsf(pl.y - pt1);
      v_sl1 = ((d0 < 1.0f) ? 0.5f * d0 * d0 : d0 - 0.5f) +
              ((d1 < 1.0f) ? 0.5f * d1 * d1 : d1 - 0.5f);
    }
    v_foc = focal_elem(prow, pcf);
  }

  v_cnt = wred_add(v_cnt);
  v_sl1 = wred_add(v_sl1);
  v_foc = wred_add(v_foc);
  if ((threadIdx.x & 31) == 0) {
    atomicAdd(&sSum[0], v_cnt);
    atomicAdd(&sSum[1], v_sl1);
    atomicAdd(&sSum[2], v_foc);
  }
  __syncthreads();
  if (threadIdx.x == 0) {
    atomicAdd(&acc[b * 8 + 5], sSum[0]);
    atomicAdd(&acc[b * 8 + 6], sSum[1]);
    atomicAdd(&acc[b * 8 + 7], sSum[2]);
  }
}

__global__ void msl_finalize(const float* __restrict__ acc, float* __restrict__ out) {
  if (threadIdx.x == 0 && blockIdx.x == 0) {
    float sl = 0.f, sc = 0.f, spl = 0.f, spc = 0.f, sct = 0.f;
    for (int b = 0; b < NBATCH; ++b) {
      float nb  = fmaxf(acc[b * 8 + 0], 1.0f);
      float pnb = fmaxf(acc[b * 8 + 5], 1.0f);
      sl  += acc[b * 8 + 2] / nb;
      sct += acc[b * 8 + 3] / nb;
      sc  += acc[b * 8 + 4] / nb;
      spl += acc[b * 8 + 6] / pnb;
      spc += acc[b * 8 + 7] / pnb;
    }
    const float inv = 1.0f / (float)NBATCH;
    out[0] = sl * inv;   // loss_l.mean()
    out[1] = sc * inv;   // loss_c.mean()
    out[2] = spl * inv;  // loss_prop_l.mean()
    out[3] = spc * inv;  // loss_prop_c.mean()
    out[4] = sct * inv;  // loss_ct.mean()
  }
}

extern "C" void kernel_launch(void* const* d_in, const int* in_sizes, int n_in,
                              void* d_out, int out_size, void* d_ws, size_t ws_size,
                              hipStream_t stream) {
  const float* loc       = (const float*)d_in[0];
  const float* conf      = (const float*)d_in[1];
  const float* prop_loc  = (const float*)d_in[2];
  const float* prop_conf = (const float*)d_in[3];
  const float* center    = (const float*)d_in[4];
  const float* priors    = (const float*)d_in[5];
  const float* targets   = (const float*)d_in[6];
  float* acc = (float*)d_ws;      // 64 floats used
  float* out = (float*)d_out;     // 5 floats

  msl_init<<<1, 64, 0, stream>>>(acc);
  dim3 grid((KANCH + 255) / 256, NBATCH);
  msl_pass1<<<grid, 256, 0, stream>>>(loc, conf, prop_loc, center, priors, targets, acc);
  msl_pass2<<<grid, 256, 0, stream>>>(loc, prop_loc, prop_conf, priors, targets, acc);
  msl_finalize<<<1, 32, 0, stream>>>(acc, out);
}